// BasePairingAttention_76115410420415
// MI455X (gfx1250) — compile-verified
//
#include <hip/hip_runtime.h>
#include <hip/hip_bf16.h>

// BasePairingAttention for MI455X (gfx1250, wave32, WMMA).
// B=8, H=8, T=S=1024, E=512, D=64, fp32.
//
// d_out layout: [0 .. 4194304)        = out  [B,T,E]
//               [4194304 .. 71303168) = attn_probs [B,H,T,S]
// d_ws layout:  qh [B,H,S,64], kh [B,H,S,64], vt [B,H,64,S] (16MB each)
//               + ctx [B,T,E] (16MB) = 64MB.

typedef __attribute__((ext_vector_type(2))) float v2f;
typedef __attribute__((ext_vector_type(4))) float v4f;
typedef __attribute__((ext_vector_type(8))) float v8f;

__device__ __forceinline__ v8f wmma4(v2f a, v2f b, v8f c) {
  // V_WMMA_F32_16X16X4_F32 : D(16x16 f32) = A(16x4 f32) * B(4x16 f32) + C
  return __builtin_amdgcn_wmma_f32_16x16x4_f32(
      /*neg_a=*/false, a, /*neg_b=*/false, b,
      /*c_mod=*/(short)0, c, /*reuse_a=*/false, /*reuse_b=*/false);
}

// -------------------------------------------------------------------------
// 512-K GEMM with bias: dst = X[M,512] @ W[512,512] + bias.
// One wave per 16x16 output tile.
//   layout==0 : flat [M,512]
//   layout==1 : head-split [B,H,row,64]          (q, k)
//   layout==2 : head-split transposed [B,H,64,S] (v) -- per-lane b128 stores
// -------------------------------------------------------------------------
__global__ __launch_bounds__(128) void gemm512_kernel(
    const float* __restrict__ X, const float* __restrict__ W,
    const float* __restrict__ bias, float* __restrict__ dst, int layout) {
  const int lane = threadIdx.x;               // 0..31
  const int wave = threadIdx.y;               // 0..3
  const int mtile = blockIdx.x;               // 0..511  (M = 8192)
  const int ntile = blockIdx.y * 4 + wave;    // 0..31   (N = 512)
  const int m0 = mtile * 16, n0 = ntile * 16;

  const int arow = lane & 15;                 // A-frag row
  const int koff = (lane >> 4) << 1;          // A/B-frag K sub-offset {0,2}
  const int bcol = lane & 15;                 // B/C-frag column
  const int chi  = (lane >> 4) << 3;          // C row offset {0,8}

  v8f c = {};
  const float* Xrow = X + (m0 + arow) * 512;
  const float* Wcol = W + n0 + bcol;
  for (int k0 = 0; k0 < 512; k0 += 32) {
    __builtin_prefetch(Xrow + k0 + 32, 0, 3);   // global_prefetch next chunk
#pragma unroll
    for (int k = k0; k < k0 + 32; k += 4) {
      v2f a = *(const v2f*)(Xrow + k + koff);         // A[row][k+koff .. +1]
      v2f b;
      b.x = Wcol[(k + koff) * 512];                   // B[k+koff  ][col]
      b.y = Wcol[(k + koff + 1) * 512];               // B[k+koff+1][col]
      c = wmma4(a, b, c);
    }
  }
  const float bn = bias[n0 + bcol];

  if (layout == 0) {
#pragma unroll
    for (int r = 0; r < 8; ++r)
      dst[(m0 + r + chi) * 512 + n0 + bcol] = c[r] + bn;
  } else if (layout == 1) {
    const int n = n0 + bcol, h = n >> 6, d = n & 63;
#pragma unroll
    for (int r = 0; r < 8; ++r) {
      const int m = m0 + r + chi;
      const int bb = m >> 10, t = m & 1023;
      dst[(((bb * 8 + h) * 1024) + t) * 64 + d] = c[r] + bn;
    }
  } else {  // transposed head-split: [B,H,64,1024]; 8 consecutive t per lane
    const int n = n0 + bcol, h = n >> 6, d = n & 63;
    const int mbase = m0 + chi;               // multiple of 8; no 1024-cross
    const int bb = mbase >> 10, tt = mbase & 1023;
    float* p = dst + ((size_t)(bb * 8 + h) * 64 + d) * 1024 + tt;
    v4f w0 = {c[0] + bn, c[1] + bn, c[2] + bn, c[3] + bn};
    v4f w1 = {c[4] + bn, c[5] + bn, c[6] + bn, c[7] + bn};
    *(v4f*)p = w0;
    *(v4f*)(p + 4) = w1;
  }
}

// -------------------------------------------------------------------------
// Fused attention: scores (WMMA) + pairing bonus + mask + softmax + P@V.
// One block = one (b,h) head x 16 query rows. 16x1024 logits staged in LDS
// (stride padded to 1028 floats -> conflict-free phase-3 A reads).
// -------------------------------------------------------------------------
#define SSTR 1028

__global__ __launch_bounds__(256) void attn_fused_kernel(
    const float* __restrict__ qh, const float* __restrict__ kh,
    const float* __restrict__ vt, const int* __restrict__ seq,
    const unsigned char* __restrict__ pad, float* __restrict__ probs,
    float* __restrict__ ctx) {
  __shared__ float sc[16 * SSTR];   // 65,792 B of the 320 KB WGP LDS

  const int tid  = threadIdx.x;
  const int lane = tid & 31, wave = tid >> 5;     // 8 waves
  const int ttile = blockIdx.x;                   // 0..63
  const int bh    = blockIdx.y;                   // 0..63
  const int b = bh >> 3, h = bh & 7;
  const int t0 = ttile * 16;

  const int arow = lane & 15;
  const int koff = (lane >> 4) << 1;
  const int bcol = lane & 15;
  const int chi  = (lane >> 4) << 3;

  // ---- Phase 1: scores = 0.125 * q @ k^T, 8 s-tiles per wave ----
  const float* qrow = qh + (bh * 1024 + t0 + arow) * 64 + koff;
  const float* kbase = kh + bh * 1024 * 64;
  for (int i = 0; i < 8; ++i) {
    const int s0 = (wave * 8 + i) * 16;
    const float* krow = kbase + (s0 + bcol) * 64 + koff;
    v8f c = {};
#pragma unroll
    for (int k = 0; k < 64; k += 4) {
      v2f a = *(const v2f*)(qrow + k);   // q[t][k..k+1]
      v2f bb = *(const v2f*)(krow + k);  // k[s][k..k+1] == (k^T)[k][s]
      c = wmma4(a, bb, c);
    }
#pragma unroll
    for (int r = 0; r < 8; ++r)
      sc[(r + chi) * SSTR + s0 + bcol] = c[r] * 0.125f;  // D^-0.5
  }
  __syncthreads();

  // ---- Phase 1.5: pairing bonus + separation + padding mask ----
  // PAIRING_MATRIX[a][b] != 0  <=>  |a-b|==1 && a<4 && b<4  (value 1.0)
  {
    const int* seqb = seq + b * 1024;
    const unsigned char* padb = pad + b * 1024;
    for (int idx = tid; idx < 16 * 1024; idx += 256) {
      const int m = idx >> 10, s = idx & 1023;
      const int t = t0 + m;
      const int na = seqb[t], nb = seqb[s];
      int dn = na - nb; dn = dn < 0 ? -dn : dn;
      int sep = t - s;  sep = sep < 0 ? -sep : sep;
      float bonus =
          (dn == 1 && na < 4 && nb < 4 && sep >= 3) ? 2.0f : 0.0f;  // PAIR_BONUS
      float v = sc[m * SSTR + s] + bonus;
      if (padb[s]) v = -__builtin_inff();
      sc[m * SSTR + s] = v;
    }
  }
  __syncthreads();

  // ---- Phase 2: row softmax (wave32 shuffle reductions), b128 probs write
  float* prow_base = probs + ((size_t)bh * 1024 + t0) * 1024;
  for (int rr = 0; rr < 2; ++rr) {
    const int m = wave * 2 + rr;
    float* srow = &sc[m * SSTR];
    float mx = -__builtin_inff();
    for (int j = lane * 4; j < 1024; j += 128) {
      v4f v = *(const v4f*)(srow + j);
      mx = fmaxf(fmaxf(fmaxf(mx, v.x), v.y), fmaxf(v.z, v.w));
    }
#pragma unroll
    for (int off = 16; off > 0; off >>= 1)
      mx = fmaxf(mx, __shfl_xor(mx, off, 32));
    float sum = 0.0f;
    for (int j = lane * 4; j < 1024; j += 128) {
      v4f v = *(const v4f*)(srow + j);
      v.x = __expf(v.x - mx); v.y = __expf(v.y - mx);
      v.z = __expf(v.z - mx); v.w = __expf(v.w - mx);
      *(v4f*)(srow + j) = v;
      sum += (v.x + v.y) + (v.z + v.w);
    }
#pragma unroll
    for (int off = 16; off > 0; off >>= 1)
      sum += __shfl_xor(sum, off, 32);
    const float inv = 1.0f / sum;
    float* prow = prow_base + m * 1024;
    for (int j = lane * 4; j < 1024; j += 128) {   // global_store_b128
      v4f v = *(const v4f*)(srow + j);
      v.x *= inv; v.y *= inv; v.z *= inv; v.w *= inv;
      *(v4f*)(srow + j) = v;
      *(v4f*)(prow + j) = v;
    }
  }
  __syncthreads();

  // ---- Phase 3: ctx(16x64) = P(16x1024) @ V(1024x64), split-K over 8 waves
  // V is transposed [64][1024] -> B-fragment pair is one contiguous b64 load.
  const float* vtb = vt + (size_t)bh * 64 * 1024;
  const int ntile = wave & 3;     // which 16-wide slice of D=64
  const int khalf = wave >> 2;    // K half: [0,512) or [512,1024)
  const int d0 = ntile * 16;
  v8f c = {};
  const float* vrow = vtb + (d0 + bcol) * 1024 + koff;
  const int kbeg = khalf * 512, kend = kbeg + 512;
  for (int k = kbeg; k < kend; k += 4) {
    v2f a = *(const v2f*)(&sc[arow * SSTR + k + koff]);  // P[t][k..k+1]
    v2f bb = *(const v2f*)(vrow + k);                    // V[k..k+1][d]
    c = wmma4(a, bb, c);
  }
  __syncthreads();                 // all P reads from LDS done; safe to reuse
  if (khalf == 1) {                // upper-half waves deposit partials
#pragma unroll
    for (int r = 0; r < 8; ++r)
      sc[(ntile * 16 + r + chi) * 16 + bcol] = c[r];
  }
  __syncthreads();
  if (khalf == 0) {                // lower-half waves combine + write ctx
    float* crow = ctx + (size_t)(b * 1024 + t0) * 512 + h * 64 + d0 + bcol;
#pragma unroll
    for (int r = 0; r < 8; ++r)
      crow[(r + chi) * 512] = c[r] + sc[(ntile * 16 + r + chi) * 16 + bcol];
  }
}

// -------------------------------------------------------------------------
extern "C" void kernel_launch(void* const* d_in, const int* in_sizes, int n_in,
                              void* d_out, int out_size, void* d_ws,
                              size_t ws_size, hipStream_t stream) {
  const float* query = (const float*)d_in[0];
  const float* key   = (const float*)d_in[1];
  const float* value = (const float*)d_in[2];
  const int*   seq   = (const int*)d_in[3];
  const unsigned char* pad = (const unsigned char*)d_in[4];
  const float* Wq = (const float*)d_in[5];  const float* bq = (const float*)d_in[6];
  const float* Wk = (const float*)d_in[7];  const float* bk = (const float*)d_in[8];
  const float* Wv = (const float*)d_in[9];  const float* bv = (const float*)d_in[10];
  const float* Wo = (const float*)d_in[11]; const float* bo = (const float*)d_in[12];

  float* out   = (float*)d_out;             // [8,1024,512]
  float* probs = out + 4194304;             // [8,8,1024,1024]

  float* qh  = (float*)d_ws;                // [8,8,1024,64]
  float* kh  = qh + 4194304;                // [8,8,1024,64]
  float* vt  = kh + 4194304;                // [8,8,64,1024] (transposed)
  float* ctx = vt + 4194304;                // [8192,512]

  const dim3 gblk(32, 4);
  const dim3 ggrid(512, 8);
  gemm512_kernel<<<ggrid, gblk, 0, stream>>>(query, Wq, bq, qh, 1);
  gemm512_kernel<<<ggrid, gblk, 0, stream>>>(key,   Wk, bk, kh, 1);
  gemm512_kernel<<<ggrid, gblk, 0, stream>>>(value, Wv, bv, vt, 2);

  attn_fused_kernel<<<dim3(64, 64), 256, 0, stream>>>(qh, kh, vt, seq, pad,
                                                      probs, ctx);

  gemm512_kernel<<<ggrid, gblk, 0, stream>>>(ctx, Wo, bo, out, 0);
}